// RGCN_55078660604130
// MI455X (gfx1250) — compile-verified
//
#include <hip/hip_runtime.h>
#include <hip/hip_bf16.h>
#include <math.h>

// ---------------------------------------------------------------------------
// RGCN forward for MI455X (gfx1250, wave32).
//   - per-relation mean aggregation via global f32 atomics (L2-resident)
//   - all dense GEMMs via V_WMMA_F32_16X16X4_F32 (exact fp32), guard-free
//     hot loop: A via b64 loads from zero-padded buffers, B pair-packed.
// ---------------------------------------------------------------------------

typedef float v2f __attribute__((ext_vector_type(2)));
typedef float v8f __attribute__((ext_vector_type(8)));

#define N_NODES   100000
#define N_EDGES   3200000
#define N_GRAPHS  256
#define N_REL     4
#define N_BASES   30
#define F_IN      86
#define HID       56
#define N_CLASSES 18

static inline int ceil_div(int a, int b) { return (a + b - 1) / b; }

// B packing: element (k, c) of a [K_pad][64] matrix lives at
//   (k>>1)*128 + c*2 + (k&1)
// so (B[k][c], B[k+1][c]) for even k is a contiguous, 8B-aligned float2.

// ---------------------------------------------------------------------------
__global__ void zero_kernel(float* __restrict__ p, size_t n) {
    size_t i = (size_t)blockIdx.x * blockDim.x + threadIdx.x;
    if (i < n) p[i] = 0.0f;
}

// inv[i] = 1 / max(cnt[i], 1)
__global__ void inv_kernel(const float* __restrict__ cnt, float* __restrict__ inv, int n) {
    int i = blockIdx.x * blockDim.x + threadIdx.x;
    if (i < n) inv[i] = 1.0f / fmaxf(cnt[i], 1.0f);
}

// copy [n_rows][F] into zero-padded [n_rows][F_pad]
__global__ void pad_rows_kernel(const float* __restrict__ X, float* __restrict__ Xp,
                                int F, int F_pad, size_t n_rows) {
    size_t i = (size_t)blockIdx.x * blockDim.x + threadIdx.x;
    size_t total = n_rows * (size_t)F_pad;
    if (i >= total) return;
    int k = (int)(i % F_pad);
    size_t row = i / F_pad;
    Xp[i] = (k < F) ? X[row * F + k] : 0.0f;
}

// ---------------------------------------------------------------------------
// W_r = sum_b comp[r,b] * basis[b], written pair-packed + zero-padded:
//   Wpad[r * K_pad * 64 + (k>>1)*128 + c*2 + (k&1)]
__global__ void combine_w_kernel(const float* __restrict__ basis, // [NB][K][C]
                                 const float* __restrict__ comp,  // [R][NB]
                                 float* __restrict__ Wpad,
                                 int R, int NB, int K, int C, int K_pad) {
    int idx = blockIdx.x * blockDim.x + threadIdx.x;
    int total = R * K_pad * 64;
    if (idx >= total) return;
    int c = idx & 63;
    int k = (idx >> 6) % K_pad;
    int r = idx / (K_pad * 64);
    float v = 0.0f;
    if (k < K && c < C) {
        for (int b = 0; b < NB; ++b)
            v += comp[r * NB + b] * basis[((size_t)b * K + k) * C + c];
    }
    Wpad[(size_t)r * K_pad * 64 + ((k >> 1) * 128 + c * 2 + (k & 1))] = v;
}

// pad a [K][C] matrix into pair-packed zero-padded [K_pad][64]
__global__ void pad_mat_kernel(const float* __restrict__ M, float* __restrict__ Mpad,
                               int K, int C, int K_pad) {
    int idx = blockIdx.x * blockDim.x + threadIdx.x;
    int total = K_pad * 64;
    if (idx >= total) return;
    int c = idx & 63;
    int k = idx >> 6;
    float v = (k < K && c < C) ? M[(size_t)k * C + c] : 0.0f;
    Mpad[(k >> 1) * 128 + c * 2 + (k & 1)] = v;
}

// ---------------------------------------------------------------------------
// Edge scatter for one relation: wave per edge, lanes cover features.
__global__ void edge_scatter_kernel(const int* __restrict__ src,
                                    const int* __restrict__ dst,
                                    const int* __restrict__ etype, int rel,
                                    const float* __restrict__ X, int F, int ldx,
                                    float* __restrict__ accum, int lda,
                                    float* __restrict__ cnt, int n_edges) {
    int wpb = blockDim.x >> 5;
    int e = blockIdx.x * wpb + (threadIdx.x >> 5);
    if (e >= n_edges) return;
    if (etype[e] != rel) return;
    int lane = threadIdx.x & 31;
    int s = src[e];
    int d = dst[e];
    for (int f = lane; f < F; f += 32)
        atomicAdd(&accum[(size_t)d * lda + f], X[(size_t)s * ldx + f]);
    if (lane == 0) atomicAdd(&cnt[d], 1.0f);
}

// ---------------------------------------------------------------------------
// WMMA fp32 GEMM: OUT[16-row tile, 56] (+)= (A * scale_row) @ B
//   A:  [nrows][lda], lda even, zero-padded so no k guard needed
//   Bp: pair-packed [K_pad/2][64][2], zero-padded
//   One wave per 16-row tile; 4 N-tiles of 16 (cols 0..63, store guard col<56)
__global__ void wmma_mm_kernel(const float* __restrict__ A, int lda,
                               const float* __restrict__ scale,   // per-row or null
                               const float* __restrict__ Bp,
                               const float* __restrict__ bias,    // [56] or null
                               float* __restrict__ OUT,           // [nrows][56]
                               int accumulate, int apply_relu, int K_pad, int ntiles) {
    int tile = blockIdx.x * (blockDim.x >> 5) + (threadIdx.x >> 5);
    if (tile >= ntiles) return;                 // wave-uniform -> EXEC all ones below
    int lane = threadIdx.x & 31;
    int half = lane >> 4;                       // 0: lanes 0-15, 1: lanes 16-31
    int ml   = lane & 15;
    int m0   = tile * 16;
    int rowA = m0 + ml;
    float sc = (scale != nullptr) ? scale[rowA] : 1.0f;

    v8f c[4];
#pragma unroll
    for (int t = 0; t < 4; ++t) {
        int col = t * 16 + ml;
#pragma unroll
        for (int j = 0; j < 8; ++j) {
            int row = m0 + j + half * 8;
            float v = 0.0f;
            if (col < HID) {
                if (accumulate) v = OUT[(size_t)row * HID + col];
                else if (bias)  v = bias[col];
            }
            c[t][j] = v;
        }
    }

    const float* arow = A + (size_t)rowA * lda;
    for (int k0 = 0; k0 < K_pad; k0 += 4) {
        int ka = k0 + half * 2;
        v2f a = *(const v2f*)(arow + ka);
        a.x *= sc;
        a.y *= sc;
        const float* bbase = Bp + (size_t)(ka >> 1) * 128 + ml * 2;
#pragma unroll
        for (int t = 0; t < 4; ++t) {
            v2f b = *(const v2f*)(bbase + t * 32);
            c[t] = __builtin_amdgcn_wmma_f32_16x16x4_f32(
                false, a, false, b, (short)0, c[t], false, false);
        }
    }

#pragma unroll
    for (int t = 0; t < 4; ++t) {
        int col = t * 16 + ml;
        if (col < HID) {
#pragma unroll
            for (int j = 0; j < 8; ++j) {
                int row = m0 + j + half * 8;
                float v = c[t][j];
                if (apply_relu) v = fmaxf(v, 0.0f);
                OUT[(size_t)row * HID + col] = v;
            }
        }
    }
}

// ---------------------------------------------------------------------------
// global_add_pool: wave per node, atomics into g[batch[node]]
__global__ void pool_kernel(const float* __restrict__ H, const int* __restrict__ batch,
                            float* __restrict__ g, int n) {
    int node = blockIdx.x * (blockDim.x >> 5) + (threadIdx.x >> 5);
    if (node >= n) return;
    int lane = threadIdx.x & 31;
    int b = batch[node];
    for (int f = lane; f < HID; f += 32)
        atomicAdd(&g[(size_t)b * HID + f], H[(size_t)node * HID + f]);
}

// ---------------------------------------------------------------------------
// BN(eval) -> fc1 -> relu -> fc2 -> log_softmax.  One thread per graph (tiny).
__global__ void head_kernel(const float* __restrict__ g,
                            const float* __restrict__ bn_gamma, const float* __restrict__ bn_beta,
                            const float* __restrict__ bn_mean,  const float* __restrict__ bn_var,
                            const float* __restrict__ fc1_w, const float* __restrict__ fc1_b,
                            const float* __restrict__ fc2_w, const float* __restrict__ fc2_b,
                            float* __restrict__ out) {
    int gi = blockIdx.x * blockDim.x + threadIdx.x;
    if (gi >= N_GRAPHS) return;
    float v[HID];
    for (int i = 0; i < HID; ++i)
        v[i] = (g[(size_t)gi * HID + i] - bn_mean[i]) * rsqrtf(bn_var[i] + 1e-5f) * bn_gamma[i]
               + bn_beta[i];
    float h[HID];
    for (int o = 0; o < HID; ++o) {
        float s = fc1_b[o];
        for (int i = 0; i < HID; ++i) s += v[i] * fc1_w[(size_t)i * HID + o];
        h[o] = fmaxf(s, 0.0f);
    }
    float logits[N_CLASSES];
    float m = -1e30f;
    for (int o = 0; o < N_CLASSES; ++o) {
        float s = fc2_b[o];
        for (int i = 0; i < HID; ++i) s += h[i] * fc2_w[(size_t)i * N_CLASSES + o];
        logits[o] = s;
        m = fmaxf(m, s);
    }
    float sum = 0.0f;
    for (int o = 0; o < N_CLASSES; ++o) sum += expf(logits[o] - m);
    float lse = logf(sum);
    for (int o = 0; o < N_CLASSES; ++o)
        out[(size_t)gi * N_CLASSES + o] = logits[o] - m - lse;
}

// ---------------------------------------------------------------------------
extern "C" void kernel_launch(void* const* d_in, const int* in_sizes, int n_in,
                              void* d_out, int out_size, void* d_ws, size_t ws_size,
                              hipStream_t stream) {
    const float* x      = (const float*)d_in[0];
    const int*   eidx   = (const int*)d_in[1];
    const int*   etype  = (const int*)d_in[2];
    const int*   batch  = (const int*)d_in[3];
    const float* basis1 = (const float*)d_in[4];
    const float* comp1  = (const float*)d_in[5];
    const float* root1  = (const float*)d_in[6];
    const float* bias1  = (const float*)d_in[7];
    const float* basis2 = (const float*)d_in[8];
    const float* comp2  = (const float*)d_in[9];
    const float* root2  = (const float*)d_in[10];
    const float* bias2  = (const float*)d_in[11];
    const float* bn_g   = (const float*)d_in[12];
    const float* bn_b   = (const float*)d_in[13];
    const float* bn_m   = (const float*)d_in[14];
    const float* bn_v   = (const float*)d_in[15];
    const float* fc1_w  = (const float*)d_in[16];
    const float* fc1_b  = (const float*)d_in[17];
    const float* fc2_w  = (const float*)d_in[18];
    const float* fc2_b  = (const float*)d_in[19];
    float* out = (float*)d_out;

    const int N = N_NODES, E = N_EDGES;
    const int* src = eidx;
    const int* dst = eidx + E;
    const int K1P = 88;                  // layer1 K (86) padded to mult of 4
    const int K2P = 56;                  // layer2 K, already mult of 4
    const int ntiles = N / 16;           // 6250 exactly

    // workspace layout (floats)
    float* w = (float*)d_ws;
    float* W1pad  = w; w += (size_t)N_REL * K1P * 64;
    float* root1p = w; w += (size_t)K1P * 64;
    float* W2pad  = w; w += (size_t)N_REL * K2P * 64;
    float* root2p = w; w += (size_t)K2P * 64;
    float* xpad   = w; w += (size_t)N * K1P;   // zero-padded copy of x
    float* out1   = w; w += (size_t)N * HID;
    float* out2   = w; w += (size_t)N * HID;
    float* accum  = w; w += (size_t)N * K1P;   // reused with stride 56 in layer 2
    float* cnt    = w; w += N;
    float* inv    = w; w += N;
    float* g      = w; w += (size_t)N_GRAPHS * HID;

    const int TB = 256;
    const int WPB = TB / 32;

    // ---- pack relation / root weights (pair-packed, zero-padded) ----
    combine_w_kernel<<<ceil_div(N_REL * K1P * 64, TB), TB, 0, stream>>>(
        basis1, comp1, W1pad, N_REL, N_BASES, F_IN, HID, K1P);
    pad_mat_kernel<<<ceil_div(K1P * 64, TB), TB, 0, stream>>>(root1, root1p, F_IN, HID, K1P);
    combine_w_kernel<<<ceil_div(N_REL * K2P * 64, TB), TB, 0, stream>>>(
        basis2, comp2, W2pad, N_REL, N_BASES, HID, HID, K2P);
    pad_mat_kernel<<<ceil_div(K2P * 64, TB), TB, 0, stream>>>(root2, root2p, HID, HID, K2P);

    // ---- zero-padded copy of x for guard-free WMMA A loads ----
    pad_rows_kernel<<<ceil_div(N * K1P, TB), TB, 0, stream>>>(x, xpad, F_IN, K1P, (size_t)N);

    const int mm_blocks = ceil_div(ntiles, WPB);
    const int eg_blocks = ceil_div(E, WPB);

    // ---- layer 1: out1 = relu(x @ root1 + bias1 + sum_r mean_r @ W1[r]) ----
    wmma_mm_kernel<<<mm_blocks, TB, 0, stream>>>(
        xpad, K1P, nullptr, root1p, bias1, out1, /*acc=*/0, /*relu=*/0, K1P, ntiles);
    for (int r = 0; r < N_REL; ++r) {
        zero_kernel<<<ceil_div(N * K1P, TB), TB, 0, stream>>>(accum, (size_t)N * K1P);
        zero_kernel<<<ceil_div(N, TB), TB, 0, stream>>>(cnt, (size_t)N);
        edge_scatter_kernel<<<eg_blocks, TB, 0, stream>>>(
            src, dst, etype, r, x, F_IN, F_IN, accum, K1P, cnt, E);
        inv_kernel<<<ceil_div(N, TB), TB, 0, stream>>>(cnt, inv, N);
        wmma_mm_kernel<<<mm_blocks, TB, 0, stream>>>(
            accum, K1P, inv, W1pad + (size_t)r * K1P * 64, nullptr, out1,
            /*acc=*/1, /*relu=*/(r == N_REL - 1) ? 1 : 0, K1P, ntiles);
    }

    // ---- layer 2: out2 = out1 @ root2 + bias2 + sum_r mean_r @ W2[r] ----
    wmma_mm_kernel<<<mm_blocks, TB, 0, stream>>>(
        out1, HID, nullptr, root2p, bias2, out2, /*acc=*/0, /*relu=*/0, K2P, ntiles);
    for (int r = 0; r < N_REL; ++r) {
        zero_kernel<<<ceil_div(N * K2P, TB), TB, 0, stream>>>(accum, (size_t)N * K2P);
        zero_kernel<<<ceil_div(N, TB), TB, 0, stream>>>(cnt, (size_t)N);
        edge_scatter_kernel<<<eg_blocks, TB, 0, stream>>>(
            src, dst, etype, r, out1, HID, HID, accum, K2P, cnt, E);
        inv_kernel<<<ceil_div(N, TB), TB, 0, stream>>>(cnt, inv, N);
        wmma_mm_kernel<<<mm_blocks, TB, 0, stream>>>(
            accum, K2P, inv, W2pad + (size_t)r * K2P * 64, nullptr, out2,
            /*acc=*/1, /*relu=*/0, K2P, ntiles);
    }

    // ---- pool + head ----
    zero_kernel<<<ceil_div(N_GRAPHS * HID, TB), TB, 0, stream>>>(g, (size_t)N_GRAPHS * HID);
    pool_kernel<<<ceil_div(N, WPB), TB, 0, stream>>>(out2, batch, g, N);
    head_kernel<<<1, N_GRAPHS, 0, stream>>>(g, bn_g, bn_b, bn_m, bn_v,
                                            fc1_w, fc1_b, fc2_w, fc2_b, out);
}